// PolyConvFrame_59339268161637
// MI455X (gfx1250) — compile-verified
//
#include <hip/hip_runtime.h>
#include <stdint.h>

// ---------------------------------------------------------------------------
// PolyConvFrame on MI455X (gfx1250): 3 chained SpMM levels (COO, scatter-add)
// Memory-bound (AI ~0.25 FLOP/B) -> optimize the data path, not the math path.
// CDNA5 features used:
//   * global_load_async_to_lds_b32 double-buffered staging of edge triples
//   * s_wait_asynccnt pipelined waits
//   * wave32-sized 16-lane-per-edge coalesced feature mapping
//   * native global_atomic_add_f32 (unsafeAtomicAdd) for the segment-sum
// ---------------------------------------------------------------------------

#define N_NODES   50000
#define N_EDGES   1600000
#define D_FEAT    48
#define DEPTH     3
#define OUT_ROW   (D_FEAT * (DEPTH + 1))   // 192 floats per node in output

#define TPB       256                      // 8 waves (wave32) per block
#define CHUNK     256                      // edges staged per pipeline stage

// ---- CDNA5 async copy: global -> LDS, one b32 per lane --------------------
__device__ __forceinline__ void async_ld_b32(uint32_t lds_addr, const void* gaddr) {
  // VDST = per-lane LDS byte address, VADDR = per-lane 64-bit global address
  asm volatile("global_load_async_to_lds_b32 %0, %1, off"
               :: "v"(lds_addr), "v"(gaddr) : "memory");
}
__device__ __forceinline__ void wait_async_0() {
  asm volatile("s_wait_asynccnt 0x0" ::: "memory");
}
__device__ __forceinline__ void wait_async_3() {
  asm volatile("s_wait_asynccnt 0x3" ::: "memory");
}

// Low 32 bits of a generic pointer to __shared__ == LDS byte offset (aperture
// check discards the upper bits; see CDNA5 ISA aperture table).
__device__ __forceinline__ uint32_t lds_off(const void* p) {
  return (uint32_t)(uintptr_t)p;
}

// ---------------------------------------------------------------------------
// Kernel 0: out[n][0][f] = x[n][f]; out[n][1..3][f] = 0   (float4 stores)
// ---------------------------------------------------------------------------
__global__ void __launch_bounds__(TPB)
poly_init_kernel(const float* __restrict__ x, float* __restrict__ out) {
  const int qpr   = OUT_ROW / 4;                 // 48 float4 per output row
  const int total = N_NODES * qpr;
  int idx = blockIdx.x * blockDim.x + threadIdx.x;
  if (idx >= total) return;
  int n = idx / qpr;
  int q = idx - n * qpr;
  float4 v = make_float4(0.f, 0.f, 0.f, 0.f);
  if (q < D_FEAT / 4) {
    v = ((const float4*)x)[(size_t)n * (D_FEAT / 4) + q];
  }
  ((float4*)out)[idx] = v;
}

// ---------------------------------------------------------------------------
// Kernel L (L = 1..3):
//   out[:, L, :] += tanh(alpha[L]) * w[e] * out[src[e], L-1, :]  scattered to dst[e]
// Persistent blocks; edge triples staged into LDS via async DMA, double-buffered.
// ---------------------------------------------------------------------------
__global__ void __launch_bounds__(TPB)
poly_spmm_kernel(const int*   __restrict__ edge_src,
                 const int*   __restrict__ edge_dst,
                 const float* __restrict__ edge_w,
                 const float* __restrict__ alpha_params,
                 float*       __restrict__ out,
                 int level, int nchunks) {
  __shared__ uint32_t s_src[2][CHUNK];
  __shared__ uint32_t s_dst[2][CHUNK];
  __shared__ float    s_w  [2][CHUNK];

  const int tid = threadIdx.x;

  // Chunk range for this persistent block (contiguous for edge-stream locality)
  const int cpb     = (nchunks + gridDim.x - 1) / gridDim.x;
  const int c_begin = blockIdx.x * cpb;
  const int c_end   = min(nchunks, c_begin + cpb);
  if (c_begin >= c_end) return;

  const float alpha = tanhf(alpha_params[level]);   // BASEALPHA = 1.0

  const int   g    = tid >> 4;      // edge group within block (0..15)
  const int   lane = tid & 15;      // feature lane (3 feats: lane, lane+16, lane+32)

  // --- stage(c, buf): unguarded (clamped) -> exactly 3 async instr per wave ---
  auto stage = [&](int c, int buf) {
    long e = (long)c * CHUNK + tid;
    if (e > (long)N_EDGES - 1) e = (long)N_EDGES - 1;   // clamp: safe reads
    async_ld_b32(lds_off(&s_src[buf][tid]), edge_src + e);
    async_ld_b32(lds_off(&s_dst[buf][tid]), edge_dst + e);
    async_ld_b32(lds_off(&s_w  [buf][tid]), edge_w   + e);
  };

  int buf = 0;
  stage(c_begin, buf);                                  // prologue

  for (int c = c_begin; c < c_end; ++c) {
    const bool has_next = (c + 1) < c_end;
    if (has_next) stage(c + 1, buf ^ 1);                // overlap next DMA
    if (has_next) wait_async_3(); else wait_async_0();  // current chunk landed
    __syncthreads();                                    // visible block-wide

    const long e_base = (long)c * CHUNK;
#pragma unroll 4
    for (int j = 0; j < CHUNK / 16; ++j) {              // 16 edges per pass
      const int  el = j * 16 + g;
      const long e  = e_base + el;
      if (e < (long)N_EDGES) {
        const int   s    = (int)s_src[buf][el];
        const int   d    = (int)s_dst[buf][el];
        const float coef = alpha * s_w[buf][el];
        const float* pin  = out + (size_t)s * OUT_ROW + (size_t)(level - 1) * D_FEAT;
        float*       pout = out + (size_t)d * OUT_ROW + (size_t)level       * D_FEAT;
#pragma unroll
        for (int k = 0; k < 3; ++k) {
          const int f = lane + 16 * k;                  // 64B-coalesced per group
          unsafeAtomicAdd(pout + f, coef * pin[f]);     // global_atomic_add_f32
        }
      }
    }
    __syncthreads();                                    // done reading buf
    buf ^= 1;
  }
}

// ---------------------------------------------------------------------------
extern "C" void kernel_launch(void* const* d_in, const int* in_sizes, int n_in,
                              void* d_out, int out_size, void* d_ws, size_t ws_size,
                              hipStream_t stream) {
  (void)in_sizes; (void)n_in; (void)out_size; (void)d_ws; (void)ws_size;

  const float* x            = (const float*)d_in[0];
  const float* edge_w       = (const float*)d_in[1];
  const float* alpha_params = (const float*)d_in[2];
  const int*   edge_src     = (const int*)  d_in[3];
  const int*   edge_dst     = (const int*)  d_in[4];
  float*       out          = (float*)d_out;

  // Level 0 + zero accumulators
  const int totalv4 = N_NODES * (OUT_ROW / 4);
  poly_init_kernel<<<(totalv4 + TPB - 1) / TPB, TPB, 0, stream>>>(x, out);

  // Three dependency-ordered SpMM levels (stream order = level barrier)
  const int nchunks = (N_EDGES + CHUNK - 1) / CHUNK;    // 6250
  const int grid    = 1250;                             // 5 chunks/block -> real double-buffering
  for (int L = 1; L <= DEPTH; ++L) {
    poly_spmm_kernel<<<grid, TPB, 0, stream>>>(edge_src, edge_dst, edge_w,
                                               alpha_params, out, L, nchunks);
  }
}